// stats_t_13297218748797
// MI455X (gfx1250) — compile-verified
//
#include <hip/hip_runtime.h>
#include <math.h>

// Problem constants (from reference setup_inputs)
#define GW     2048
#define GH     2048
#define DMAX   1024
#define NBEAM  1081
#define NSEED  8
#define NRAYS  (NBEAM * NSEED)
#define OCC_THRESH 0.999f
#define TWO_PI_F   6.283185307179586f
#define SIG_ANG_F  0.08726646259971647f   // 5 deg in radians

// Copy the (zero) counts input into the output each launch so repeated
// graph replays are deterministic (harness poisons d_out once, never restores).
__global__ void init_counts_kernel(const float* __restrict__ cin,
                                   float* __restrict__ cout, int n) {
  int i = blockIdx.x * blockDim.x + threadIdx.x;
  if (i < n) cout[i] = cin[i];
}

// One wave32 per ray. Lanes test 32 consecutive step indices at once;
// ballot_w32 + ctz gives the first hit; wave early-exits on any hit.
// Depth-2 software pipeline on the gathers + global_prefetch_b8 at distance 4.
__global__ __launch_bounds__(256) void raymarch_kernel(
    const float* __restrict__ grid,     // [GH,GW]
    const float* __restrict__ pos,      // [2]
    const float* __restrict__ rot,      // [1]
    const float* __restrict__ angles,   // [NBEAM]
    const float* __restrict__ dists,    // [NBEAM]
    const float* __restrict__ snoise,   // [NSEED,NBEAM,2] flattened
    const float* __restrict__ anoise,   // [NSEED,NBEAM]   flattened
    float* __restrict__ counts)         // [DMAX,DMAX]
{
  const int lane = threadIdx.x & 31;
  const int ray  = blockIdx.x * (blockDim.x >> 5) + (threadIdx.x >> 5);
  if (ray >= NRAYS) return;

  const int s = ray / NBEAM;
  const int b = ray - s * NBEAM;
  (void)s;

  // Noisy origin (pixel coords) and heading — matches reference exactly.
  const float sx = fmaf(pos[0], (float)GW, snoise[2 * ray + 0]);  // SIG_POS == 1.0
  const float sy = fmaf(pos[1], (float)GH, snoise[2 * ray + 1]);
  const float th = rot[0] + angles[b] * TWO_PI_F + anoise[ray] * SIG_ANG_F;
  float dy, dx;
  sincosf(th, &dy, &dx);   // dx = cos(theta), dy = sin(theta)

  const float t0 = (float)lane;

  // Address of the grid cell this lane samples in chunk k (steps k*32+lane).
  auto cell = [&](int k) -> const float* {
    float t  = t0 + (float)(k << 5);
    float px = fmaf(t, dx, sx);
    float py = fmaf(t, dy, sy);
    int xi = (int)floorf(px);
    int yi = (int)floorf(py);
    xi = xi < 0 ? 0 : (xi > GW - 1 ? GW - 1 : xi);
    yi = yi < 0 ? 0 : (yi > GH - 1 ? GH - 1 : yi);
    return grid + ((yi << 11) + xi);   // GW == 2048 == 1<<11
  };

  const int NCHUNK = DMAX / 32;        // 32 chunks of 32 steps
  int firstHit = DMAX;                 // "no hit" sentinel, matches max_steps

  float v0 = *cell(0);                 // pipeline prologue (2 chunks in flight)
  float v1 = *cell(1);
  for (int k = 0; k < NCHUNK; ++k) {
    if (k + 4 < NCHUNK)
      __builtin_prefetch(cell(k + 4), 0, 3);        // -> global_prefetch_b8
    float v2 = (k + 2 < NCHUNK) ? *cell(k + 2) : 0.0f;
    unsigned hitmask = __builtin_amdgcn_ballot_w32(v0 > OCC_THRESH);
    if (hitmask) {                      // uniform across the wave
      firstHit = (k << 5) + __builtin_ctz(hitmask); // == argmax of first chunk w/ hit
      break;
    }
    v0 = v1;
    v1 = v2;
  }

  if (lane == 0) {
    // truth index: round_clip(lidar_dists[b]*D, 0, D) then min(D-1)  == clamp [0,1023]
    float truth = dists[b] * (float)DMAX;
    int ti = (int)rintf(truth);
    ti = ti < 0 ? 0 : (ti > DMAX - 1 ? DMAX - 1 : ti);
    // measured is integer-valued (first hit or 1024); round is exact.
    int mi = firstHit > DMAX - 1 ? DMAX - 1 : firstHit;
    atomicAdd(counts + ti * DMAX + mi, 1.0f);       // order-independent (+1.0 each)
  }
}

extern "C" void kernel_launch(void* const* d_in, const int* in_sizes, int n_in,
                              void* d_out, int out_size, void* d_ws, size_t ws_size,
                              hipStream_t stream) {
  const float* grid      = (const float*)d_in[0];
  const float* pos       = (const float*)d_in[1];
  const float* rot       = (const float*)d_in[2];
  const float* angles    = (const float*)d_in[3];
  const float* dists     = (const float*)d_in[4];
  const float* snoise    = (const float*)d_in[5];
  const float* anoise    = (const float*)d_in[6];
  const float* counts_in = (const float*)d_in[7];
  float* out = (float*)d_out;
  (void)in_sizes; (void)n_in; (void)d_ws; (void)ws_size;

  const int n = DMAX * DMAX;
  init_counts_kernel<<<(n + 255) / 256, 256, 0, stream>>>(counts_in, out, n);

  const int waves_per_block = 256 / 32;                            // 8 rays/block
  const int nblocks = (NRAYS + waves_per_block - 1) / waves_per_block; // 1081
  raymarch_kernel<<<nblocks, 256, 0, stream>>>(
      grid, pos, rot, angles, dists, snoise, anoise, out);
}